// IPR_MPNN_Conv_6021544149663
// MI455X (gfx1250) — compile-verified
//
#include <hip/hip_runtime.h>
#include <hip/hip_bf16.h>

// Problem constants (fixed by the reference's setup_inputs)
#define NROWS   800000
#define NF      128        // n_in
#define NB      8          // n_batches
#define RR      100000     // rows per batch
#define NV      2          // n_virtual
#define TOPK    4096
#define DD      256        // d = NV*NF

typedef __attribute__((ext_vector_type(2))) float v2f;
typedef __attribute__((ext_vector_type(8))) float v8f;

// Monotonic key: larger float  ->  larger unsigned key (total order, descending top-K = largest keys)
__device__ __forceinline__ unsigned fkey(float f) {
    unsigned u = __float_as_uint(f);
    return (u & 0x80000000u) ? ~u : (u | 0x80000000u);
}

// ---------------------------------------------------------------------------
// Pass 1: scores = x @ W_fr + b_fr, store as radix keys. One wave per row.
// ---------------------------------------------------------------------------
__global__ void scores_kernel(const float* __restrict__ x,
                              const float* __restrict__ Wfr,
                              const float* __restrict__ bfr,
                              unsigned* __restrict__ keys)
{
    const int lane = threadIdx.x & 31;
    const int wave = threadIdx.x >> 5;
    const int waveId = blockIdx.x * (blockDim.x >> 5) + wave;
    const int nWaves = gridDim.x * (blockDim.x >> 5);

    // Each lane owns features [lane*4, lane*4+4)
    float w00 = Wfr[(lane * 4 + 0) * 2 + 0], w01 = Wfr[(lane * 4 + 0) * 2 + 1];
    float w10 = Wfr[(lane * 4 + 1) * 2 + 0], w11 = Wfr[(lane * 4 + 1) * 2 + 1];
    float w20 = Wfr[(lane * 4 + 2) * 2 + 0], w21 = Wfr[(lane * 4 + 2) * 2 + 1];
    float w30 = Wfr[(lane * 4 + 3) * 2 + 0], w31 = Wfr[(lane * 4 + 3) * 2 + 1];
    float bf0 = bfr[0], bf1 = bfr[1];

    for (int row = waveId; row < NROWS; row += nWaves) {
        const float4 xv = ((const float4*)(x + (size_t)row * NF))[lane];
        float s0 = xv.x * w00 + xv.y * w10 + xv.z * w20 + xv.w * w30;
        float s1 = xv.x * w01 + xv.y * w11 + xv.z * w21 + xv.w * w31;
        #pragma unroll
        for (int off = 16; off > 0; off >>= 1) {
            s0 += __shfl_down(s0, off, 32);
            s1 += __shfl_down(s1, off, 32);
        }
        if (lane == 0) {
            keys[row]         = fkey(s0 + bf0);
            keys[NROWS + row] = fkey(s1 + bf1);
        }
    }
}

// ---------------------------------------------------------------------------
// Pass 2: 4-round byte radix-select of the K-th largest key per (b,v).
// 16 blocks; each scans its 100000-key slice from L2 four times.
// ---------------------------------------------------------------------------
__global__ void radix_select_kernel(const unsigned* __restrict__ keys,
                                    unsigned* __restrict__ thr)
{
    const int v = blockIdx.x & 1;
    const int b = blockIdx.x >> 1;
    const unsigned* s = keys + (size_t)v * NROWS + (size_t)b * RR;

    __shared__ unsigned hist[256];
    __shared__ unsigned prefix_s, remK_s;
    if (threadIdx.x == 0) { prefix_s = 0u; remK_s = TOPK; }
    __syncthreads();

    for (int round = 0; round < 4; ++round) {
        const int shift = 24 - round * 8;
        hist[threadIdx.x] = 0u;
        __syncthreads();
        const unsigned prefix = prefix_s;
        const unsigned pm = (round == 0) ? 0u : (0xFFFFFFFFu << (shift + 8));
        for (int i = threadIdx.x; i < RR; i += 256) {
            unsigned k = s[i];
            if ((k & pm) == prefix) atomicAdd(&hist[(k >> shift) & 0xFF], 1u);
        }
        __syncthreads();
        if (threadIdx.x == 0) {
            unsigned rem = remK_s, cum = 0;
            for (int bin = 255; bin >= 0; --bin) {
                unsigned c = hist[bin];
                if (cum + c >= rem) {
                    remK_s = rem - cum;
                    prefix_s = prefix | ((unsigned)bin << shift);
                    break;
                }
                cum += c;
            }
        }
        __syncthreads();
    }
    if (threadIdx.x == 0) thr[blockIdx.x] = prefix_s; // thr[b*2+v]
}

// ---------------------------------------------------------------------------
// Pass 3: build mask byte per row, accumulate per-block virt partial sums.
// Deterministic: per-wave register accs, fixed-order cross-wave LDS reduce.
// Each block owns a 1000-row chunk of a single batch (RR % 1000 == 0).
// ---------------------------------------------------------------------------
#define CHUNK 1000
__global__ void mask_virt_kernel(const float* __restrict__ x,
                                 const unsigned* __restrict__ keys,
                                 const unsigned* __restrict__ thr,
                                 unsigned char* __restrict__ mask,
                                 float* __restrict__ partials)
{
    const int row0 = blockIdx.x * CHUNK;
    const int b = row0 / RR;
    const unsigned t0 = thr[b * 2 + 0];
    const unsigned t1 = thr[b * 2 + 1];
    const int lane = threadIdx.x & 31;
    const int wave = threadIdx.x >> 5;

    float4 a0 = {0.f, 0.f, 0.f, 0.f};
    float4 a1 = {0.f, 0.f, 0.f, 0.f};

    for (int i = wave; i < CHUNK; i += 8) {
        const int row = row0 + i;
        const unsigned k0 = keys[row];
        const unsigned k1 = keys[NROWS + row];
        const bool m0 = (k0 >= t0);
        const bool m1 = (k1 >= t1);
        if (lane == 0) mask[row] = (unsigned char)((m0 ? 1 : 0) | (m1 ? 2 : 0));
        if (m0 | m1) {
            const float4 xv = ((const float4*)(x + (size_t)row * NF))[lane];
            if (m0) { a0.x += xv.x; a0.y += xv.y; a0.z += xv.z; a0.w += xv.w; }
            if (m1) { a1.x += xv.x; a1.y += xv.y; a1.z += xv.z; a1.w += xv.w; }
        }
    }

    __shared__ float sacc[8][2][NF];
    sacc[wave][0][lane * 4 + 0] = a0.x; sacc[wave][0][lane * 4 + 1] = a0.y;
    sacc[wave][0][lane * 4 + 2] = a0.z; sacc[wave][0][lane * 4 + 3] = a0.w;
    sacc[wave][1][lane * 4 + 0] = a1.x; sacc[wave][1][lane * 4 + 1] = a1.y;
    sacc[wave][1][lane * 4 + 2] = a1.z; sacc[wave][1][lane * 4 + 3] = a1.w;
    __syncthreads();

    // thread t handles (v,f) slot t; fixed-order sum over the 8 waves
    const int v = threadIdx.x >> 7;
    const int f = threadIdx.x & 127;
    float s = 0.f;
    #pragma unroll
    for (int w = 0; w < 8; ++w) s += sacc[w][v][f];
    partials[(size_t)blockIdx.x * 256 + threadIdx.x] = s;
}

// ---------------------------------------------------------------------------
// Pass 4: deterministic reduction of the 100 chunk-partials per batch.
// ---------------------------------------------------------------------------
__global__ void reduce_virt_kernel(const float* __restrict__ partials,
                                   float* __restrict__ virt)
{
    const int idx = blockIdx.x * blockDim.x + threadIdx.x; // 0..2047
    const int b  = idx >> 8;
    const int vf = idx & 255;
    float s = 0.f;
    for (int c = 0; c < RR / CHUNK; ++c)
        s += partials[(size_t)(b * (RR / CHUNK) + c) * 256 + vf];
    virt[idx] = s; // layout [b][v*128+f] == h0 row-major (8 x 256)
}

// ---------------------------------------------------------------------------
// Pass 5: MLP  h = relu(A@W1+b1)@W2+b2   via V_WMMA_F32_16X16X4_F32.
// One block, 256 threads = 8 waves; wave w owns columns [w*32, w*32+32).
// A is 8x256 zero-padded to 16x256 in LDS (stride 257 to avoid bank conflicts).
// ---------------------------------------------------------------------------
#define LSTR 257
__global__ void mlp_kernel(const float* __restrict__ virt,
                           const float* __restrict__ W1, const float* __restrict__ b1,
                           const float* __restrict__ W2, const float* __restrict__ b2,
                           float* __restrict__ vupd)
{
    __shared__ float A[16][LSTR];
    __shared__ float H[16][LSTR];

    for (int i = threadIdx.x; i < 16 * 256; i += 256) {
        const int r = i >> 8, c = i & 255;
        A[r][c] = (r < NB) ? virt[r * 256 + c] * (1.0f / (float)TOPK) : 0.0f;
    }
    __syncthreads();

    const int lane  = threadIdx.x & 31;
    const int wave  = threadIdx.x >> 5;
    const int mrow  = lane & 15;          // A-frag: M row (lanes 0-15 & 16-31)
    const int khalf = (lane >> 4) * 2;    // A/B-frag: K sub-offset (0 or 2)
    const int rbase = (lane < 16) ? 0 : 8; // C/D rows: VGPR i -> M = rbase+i

    // ---- layer 1: H = relu(A @ W1 + b1) ----
    #pragma unroll
    for (int t = 0; t < 2; ++t) {
        const int n0  = wave * 32 + t * 16;
        const int col = n0 + mrow;
        v8f c = {};
        for (int k = 0; k < 256; k += 4) {
            v2f a, bb;
            a.x  = A[mrow][k + khalf];
            a.y  = A[mrow][k + khalf + 1];
            bb.x = W1[(size_t)(k + khalf) * 256 + col];
            bb.y = W1[(size_t)(k + khalf + 1) * 256 + col];
            c = __builtin_amdgcn_wmma_f32_16x16x4_f32(false, a, false, bb,
                                                      (short)0, c, false, false);
        }
        const float bias = b1[col];
        #pragma unroll
        for (int i = 0; i < 8; ++i) {
            const float hv = c[i] + bias;
            H[rbase + i][col] = hv > 0.f ? hv : 0.f;
        }
    }
    __syncthreads();

    // ---- layer 2: vupd = H @ W2 + b2 (only rows 0..7 are real batches) ----
    #pragma unroll
    for (int t = 0; t < 2; ++t) {
        const int n0  = wave * 32 + t * 16;
        const int col = n0 + mrow;
        v8f c = {};
        for (int k = 0; k < 256; k += 4) {
            v2f a, bb;
            a.x  = H[mrow][k + khalf];
            a.y  = H[mrow][k + khalf + 1];
            bb.x = W2[(size_t)(k + khalf) * 256 + col];
            bb.y = W2[(size_t)(k + khalf + 1) * 256 + col];
            c = __builtin_amdgcn_wmma_f32_16x16x4_f32(false, a, false, bb,
                                                      (short)0, c, false, false);
        }
        if (lane < 16) {
            const float bias = b2[col];
            #pragma unroll
            for (int i = 0; i < 8; ++i)             // M = 0..7 = batches
                vupd[(size_t)i * 256 + col] = c[i] + bias;
        }
    }
}

// ---------------------------------------------------------------------------
// Pass 6: streaming blend. One wave per row, float4 per lane.
// out = x;  if m0: out=(out+u0)*0.5;  if m1: out=(out+u1)*0.5
// ---------------------------------------------------------------------------
__global__ void output_kernel(const float* __restrict__ x,
                              const unsigned char* __restrict__ mask,
                              const float* __restrict__ vupd,
                              float* __restrict__ out)
{
    const int lane = threadIdx.x & 31;
    const int wave = threadIdx.x >> 5;
    const int waveId = blockIdx.x * (blockDim.x >> 5) + wave;
    const int nWaves = gridDim.x * (blockDim.x >> 5);

    for (int row = waveId; row < NROWS; row += nWaves) {
        const int b = row / RR;
        const unsigned m = mask[row];
        float4 v = ((const float4*)(x + (size_t)row * NF))[lane];
        if (m & 1u) {
            const float4 u = ((const float4*)(vupd + (size_t)b * 256))[lane];
            v.x = (v.x + u.x) * 0.5f; v.y = (v.y + u.y) * 0.5f;
            v.z = (v.z + u.z) * 0.5f; v.w = (v.w + u.w) * 0.5f;
        }
        if (m & 2u) {
            const float4 u = ((const float4*)(vupd + (size_t)b * 256 + 128))[lane];
            v.x = (v.x + u.x) * 0.5f; v.y = (v.y + u.y) * 0.5f;
            v.z = (v.z + u.z) * 0.5f; v.w = (v.w + u.w) * 0.5f;
        }
        ((float4*)(out + (size_t)row * NF))[lane] = v;
    }
}

// ---------------------------------------------------------------------------
extern "C" void kernel_launch(void* const* d_in, const int* in_sizes, int n_in,
                              void* d_out, int out_size, void* d_ws, size_t ws_size,
                              hipStream_t stream) {
    const float* x   = (const float*)d_in[0];
    const float* Wfr = (const float*)d_in[1];
    const float* bfr = (const float*)d_in[2];
    const float* W1  = (const float*)d_in[3];
    const float* b1  = (const float*)d_in[4];
    const float* W2  = (const float*)d_in[5];
    const float* b2  = (const float*)d_in[6];
    float* out = (float*)d_out;

    // workspace layout (256B aligned slots)
    char* ws = (char*)d_ws;
    size_t off = 0;
    unsigned*      keys     = (unsigned*)(ws + off); off += (size_t)2 * NROWS * 4;           // 6.40 MB
    unsigned*      thr      = (unsigned*)(ws + off); off += 256;                             // 16 used
    unsigned char* mask     = (unsigned char*)(ws + off); off += (NROWS + 255) & ~255ull;    // 0.80 MB
    float*         partials = (float*)(ws + off); off += (size_t)(NROWS / CHUNK) * 256 * 4;  // 0.82 MB
    float*         virt     = (float*)(ws + off); off += (size_t)NB * 256 * 4;
    float*         vupd     = (float*)(ws + off); off += (size_t)NB * 256 * 4;

    scores_kernel<<<2048, 256, 0, stream>>>(x, Wfr, bfr, keys);
    radix_select_kernel<<<NB * NV, 256, 0, stream>>>(keys, thr);
    mask_virt_kernel<<<NROWS / CHUNK, 256, 0, stream>>>(x, keys, thr, mask, partials);
    reduce_virt_kernel<<<NB, 256, 0, stream>>>(partials, virt);
    mlp_kernel<<<1, 256, 0, stream>>>(virt, W1, b1, W2, b2, vupd);
    output_kernel<<<4096, 256, 0, stream>>>(x, mask, vupd, out);
}